// ARD_67765993997201
// MI455X (gfx1250) — compile-verified
//
#include <hip/hip_runtime.h>
#include <hip/hip_bf16.h>

typedef float v2f __attribute__((ext_vector_type(2)));
typedef float v4f __attribute__((ext_vector_type(4)));
typedef float v8f __attribute__((ext_vector_type(8)));

#define LDSTRIDE 132   // 128 + 4 pad: avoids 16-way LDS bank conflicts on column reads

// ---------------- WMMA helpers (fp32 16x16x4) ----------------

__device__ __forceinline__ v8f wmma4(v2f a, v2f b, v8f c) {
    // 8 args: (neg_a, A, neg_b, B, c_mod, C, reuse_a, reuse_b)
    return __builtin_amdgcn_wmma_f32_16x16x4_f32(false, a, false, b, (short)0, c, false, false);
}

// One wave computes a 16(row) x 16(col) tile of  X(16x128) @ W(128x128)^T ,
// i.e. out[m][j0+n] = sum_d X[m][d] * W[j0+n][d].
// X is LDS with row stride LDSTRIDE. W is row-major global (row = output col).
__device__ __forceinline__ v8f tile_gemm(const float* X, const float* __restrict__ W,
                                         int j0, int lane) {
    v8f c = {};
    const int m    = lane & 15;
    const int half = lane >> 4;           // 0: K even pair, 1: K odd pair
    const float* wrow = W + (size_t)(j0 + (lane & 15)) * 128;
    #pragma unroll
    for (int k0 = 0; k0 < 128; k0 += 4) {
        v2f a, b;
        const int ka = k0 + half * 2;
        a.x = X[m * LDSTRIDE + ka];
        a.y = X[m * LDSTRIDE + ka + 1];
        b.x = wrow[ka];
        b.y = wrow[ka + 1];
        c = wmma4(a, b, c);
    }
    return c;
}

// Load a 16x128 fp32 tile (row stride 128 in global) into LDS (row stride LDSTRIDE).
__device__ __forceinline__ void load_tile16(const float* __restrict__ src, float* X, int tid) {
    for (int i = tid; i < 512; i += 256) {          // 512 float4s
        const int row = i >> 5, c4 = (i & 31) * 4;
        v4f v = *(const v4f*)(src + (size_t)row * 128 + c4);
        float* d = X + row * LDSTRIDE + c4;
        d[0] = v.x; d[1] = v.y; d[2] = v.z; d[3] = v.w;
    }
}

// LayerNorm over 16 rows of 128 held in LDS Y (stride LDSTRIDE), in-place.
// Optionally also writes to gout (row stride 128; global or LDS).
__device__ __forceinline__ void ln_rows(float* Y, const float* __restrict__ g,
                                        const float* __restrict__ bta, float* gout) {
    const int tid = threadIdx.x;
    const int w = tid >> 5, l = tid & 31;
    const int row = 2 * w + (l >> 4);     // 8 waves x 2 rows = 16 rows
    const int cl  = (l & 15) * 8;         // 16 lanes x 8 cols = 128 cols
    float* yr = Y + row * LDSTRIDE + cl;
    float x[8];
    float sum = 0.f;
    #pragma unroll
    for (int j = 0; j < 8; ++j) { x[j] = yr[j]; sum += x[j]; }
    #pragma unroll
    for (int off = 1; off < 16; off <<= 1) sum += __shfl_xor(sum, off);
    const float mu = sum * (1.0f / 128.0f);
    float vs = 0.f;
    #pragma unroll
    for (int j = 0; j < 8; ++j) { const float d = x[j] - mu; vs += d * d; }
    #pragma unroll
    for (int off = 1; off < 16; off <<= 1) vs += __shfl_xor(vs, off);
    const float rs = rsqrtf(vs * (1.0f / 128.0f) + 1e-5f);
    #pragma unroll
    for (int j = 0; j < 8; ++j) {
        const float o = (x[j] - mu) * rs * g[cl + j] + bta[cl + j];
        yr[j] = o;
        if (gout) gout[(size_t)row * 128 + cl + j] = o;
    }
}

// ---------------- Kernel A: q/k/v projections (WMMA) ----------------

__global__ void ARD_qkv_kernel(const float* __restrict__ ht, const float* __restrict__ W,
                               const float* __restrict__ bias,
                               float* __restrict__ qw, float* __restrict__ kw,
                               float* __restrict__ vw) {
    __shared__ float X[16 * LDSTRIDE];
    const int g = blockIdx.x, tid = threadIdx.x;
    load_tile16(ht + (size_t)g * 16 * 128, X, tid);
    __syncthreads();
    const int lane = tid & 31, wv = tid >> 5;
    const int j0 = wv * 16, half = lane >> 4, col = j0 + (lane & 15);
    float* outs[3] = {qw, kw, vw};
    for (int li = 0; li < 3; ++li) {
        v8f c = tile_gemm(X, W + (size_t)li * 128 * 128, j0, lane);
        const float* bi = bias + li * 128;
        float* dst = outs[li] + (size_t)g * 16 * 128;
        #pragma unroll
        for (int r = 0; r < 8; ++r) {
            const int row = r + half * 8;
            dst[(size_t)row * 128 + col] = c[r] + bi[col];
        }
    }
}

// ---------------- Attention kernel (flash-style online softmax) ----------------
// One block per batch row; 8 waves split the key range. Per wave-iteration a wave
// cooperatively loads one 512B key row (perfectly coalesced): lane l handles dims
// 4l..4l+3 (head = l/4); shfl_xor(1),(2) reduces the dot within each head group.
__global__ void ARD_attn_kernel(const float* __restrict__ q, const float* __restrict__ K,
                                const float* __restrict__ V, const float* __restrict__ Kx,
                                const float* __restrict__ Vx, const int* __restrict__ mask,
                                int nMain, int nTot, float* __restrict__ out) {
    const int b = blockIdx.x, tid = threadIdx.x;
    const int wv = tid >> 5, lane = tid & 31;
    __shared__ float lm[8][8];        // per-wave per-head running max
    __shared__ float lssum[8][8];     // per-wave per-head running denom
    __shared__ float lo[8][128];      // per-wave output accumulator

    const v4f q4 = *(const v4f*)(q + (size_t)b * 128 + lane * 4);
    float mi = -INFINITY, si = 0.f;
    v4f acc = {0.f, 0.f, 0.f, 0.f};
    const size_t kvbase = (size_t)b * nMain * 128;

    #pragma unroll 2
    for (int n = wv; n < nTot; n += 8) {
        const float *kp, *vp;
        if (n < nMain) { kp = K + kvbase + (size_t)n * 128; vp = V + kvbase + (size_t)n * 128; }
        else           { kp = Kx + (size_t)b * 128;         vp = Vx + (size_t)b * 128; }
        const v4f k4 = __builtin_nontemporal_load((const v4f*)(kp + lane * 4));
        const v4f v4 = __builtin_nontemporal_load((const v4f*)(vp + lane * 4));
        float s = q4.x * k4.x + q4.y * k4.y + q4.z * k4.z + q4.w * k4.w;
        s += __shfl_xor(s, 1);
        s += __shfl_xor(s, 2);        // now s = head dot for all 4 lanes of the group
        s *= 0.25f;                   // 1/sqrt(16)
        if (mask && mask[(size_t)b * nMain + n] == 1) s = -1.0e9f;
        const float mn = fmaxf(mi, s);
        const float sc = __expf(mi - mn);
        const float p  = __expf(s - mn);
        si  = si * sc + p;
        acc = acc * sc + p * v4;
        mi  = mn;
    }

    const int h = lane >> 2;
    if ((lane & 3) == 0) { lm[wv][h] = mi; lssum[wv][h] = si; }
    lo[wv][lane * 4 + 0] = acc.x;
    lo[wv][lane * 4 + 1] = acc.y;
    lo[wv][lane * 4 + 2] = acc.z;
    lo[wv][lane * 4 + 3] = acc.w;
    __syncthreads();

    if (tid < 128) {                  // merge 8 waves; head = tid/16, dim = tid
        const int hh = tid >> 4;
        float M = -INFINITY;
        #pragma unroll
        for (int w = 0; w < 8; ++w) M = fmaxf(M, lm[w][hh]);
        float den = 0.f, num = 0.f;
        #pragma unroll
        for (int w = 0; w < 8; ++w) {
            const float e = __expf(lm[w][hh] - M);
            den += lssum[w][hh] * e;
            num += lo[w][tid] * e;
        }
        out[(size_t)b * 128 + tid] = num / den;
    }
}

// ---------------- Kernel C: residual+W3, LN0, W4 query (WMMA) ----------------

__global__ void ARD_block1_kernel(const float* __restrict__ ht, const float* __restrict__ o1,
                                  const float* __restrict__ W, const float* __restrict__ bias,
                                  const float* __restrict__ lng, const float* __restrict__ lnb,
                                  float* __restrict__ ht1, float* __restrict__ query) {
    __shared__ float X[16 * LDSTRIDE];
    __shared__ float Y[16 * LDSTRIDE];
    const int g = blockIdx.x, tid = threadIdx.x;
    load_tile16(o1 + (size_t)g * 2048, X, tid);
    __syncthreads();
    const int lane = tid & 31, wv = tid >> 5;
    const int j0 = wv * 16, half = lane >> 4, col = j0 + (lane & 15);

    v8f c = tile_gemm(X, W + (size_t)3 * 128 * 128, j0, lane);
    const float* b3 = bias + 3 * 128;
    const float* htb = ht + (size_t)g * 2048;
    #pragma unroll
    for (int r = 0; r < 8; ++r) {
        const int row = r + half * 8;
        Y[row * LDSTRIDE + col] = c[r] + b3[col] + htb[(size_t)row * 128 + col];
    }
    __syncthreads();
    ln_rows(Y, lng, lnb, ht1 + (size_t)g * 2048);
    __syncthreads();

    v8f c2 = tile_gemm(Y, W + (size_t)4 * 128 * 128, j0, lane);
    const float* b4 = bias + 4 * 128;
    float* qb = query + (size_t)g * 2048;
    #pragma unroll
    for (int r = 0; r < 8; ++r) {
        const int row = r + half * 8;
        qb[(size_t)row * 128 + col] = c2[r] + b4[col];
    }
}

// ---------------- Kernel E: residual+W5, LN1, FFN(W7,relu,W6), LN2 (WMMA) ----------------

__global__ void ARD_block2_kernel(const float* __restrict__ ht1, const float* __restrict__ o2,
                                  const float* __restrict__ W, const float* __restrict__ bias,
                                  const float* __restrict__ lng, const float* __restrict__ lnb,
                                  float* __restrict__ out) {
    __shared__ float X[16 * LDSTRIDE];
    __shared__ float Y[16 * LDSTRIDE];
    __shared__ float H2[16 * 128];
    const int g = blockIdx.x, tid = threadIdx.x;
    load_tile16(o2 + (size_t)g * 2048, X, tid);
    __syncthreads();
    const int lane = tid & 31, wv = tid >> 5;
    const int j0 = wv * 16, half = lane >> 4, col = j0 + (lane & 15);

    // ht1 + lin(o2, W5, b5)
    v8f c = tile_gemm(X, W + (size_t)5 * 128 * 128, j0, lane);
    const float* b5 = bias + 5 * 128;
    const float* hb = ht1 + (size_t)g * 2048;
    #pragma unroll
    for (int r = 0; r < 8; ++r) {
        const int row = r + half * 8;
        Y[row * LDSTRIDE + col] = c[r] + b5[col] + hb[(size_t)row * 128 + col];
    }
    __syncthreads();
    ln_rows(Y, lng + 128, lnb + 128, H2);      // LN1 -> Y (and residual copy in H2)
    __syncthreads();

    // relu(lin(h2, W7, b7)) -> X
    c = tile_gemm(Y, W + (size_t)7 * 128 * 128, j0, lane);
    const float* b7 = bias + 7 * 128;
    #pragma unroll
    for (int r = 0; r < 8; ++r) {
        const int row = r + half * 8;
        X[row * LDSTRIDE + col] = fmaxf(0.f, c[r] + b7[col]);
    }
    __syncthreads();

    // h2 + lin(f, W6, b6) -> Y
    c = tile_gemm(X, W + (size_t)6 * 128 * 128, j0, lane);
    const float* b6 = bias + 6 * 128;
    #pragma unroll
    for (int r = 0; r < 8; ++r) {
        const int row = r + half * 8;
        Y[row * LDSTRIDE + col] = c[r] + b6[col] + H2[row * 128 + col];
    }
    __syncthreads();
    ln_rows(Y, lng + 256, lnb + 256, out + (size_t)g * 2048);   // LN2 -> d_out
}

// ---------------- Host launch ----------------

extern "C" void kernel_launch(void* const* d_in, const int* in_sizes, int n_in,
                              void* d_out, int out_size, void* d_ws, size_t ws_size,
                              hipStream_t stream) {
    const float* ht    = (const float*)d_in[0];   // (512,128)
    const float* key   = (const float*)d_in[1];   // (512,4096,128)
    const float* value = (const float*)d_in[2];   // (512,4096,128)
    const int*   mask  = (const int*)  d_in[3];   // (512,4096)
    const float* kprev = (const float*)d_in[4];   // (512,2048,128)
    const float* vprev = (const float*)d_in[5];   // (512,2048,128)
    const float* W     = (const float*)d_in[6];   // (8,128,128)
    const float* bias  = (const float*)d_in[7];   // (8,128)
    const float* lng   = (const float*)d_in[8];   // (3,128)
    const float* lnb   = (const float*)d_in[9];   // (3,128)

    float* ws  = (float*)d_ws;
    const size_t BD = (size_t)512 * 128;          // 65536 floats per (B,D) buffer
    float* qw  = ws + 0 * BD;
    float* kw  = ws + 1 * BD;
    float* vw  = ws + 2 * BD;
    float* o1  = ws + 3 * BD;
    float* ht1 = ws + 4 * BD;
    float* qr  = ws + 5 * BD;
    float* o2  = ws + 6 * BD;

    ARD_qkv_kernel<<<32, 256, 0, stream>>>(ht, W, bias, qw, kw, vw);
    ARD_attn_kernel<<<512, 256, 0, stream>>>(qw, kprev, vprev, kw, vw, nullptr, 2048, 2049, o1);
    ARD_block1_kernel<<<32, 256, 0, stream>>>(ht, o1, W, bias, lng, lnb, ht1, qr);
    ARD_attn_kernel<<<512, 256, 0, stream>>>(qr, key, value, nullptr, nullptr, mask, 4096, 4096, o2);
    ARD_block2_kernel<<<32, 256, 0, stream>>>(ht1, o2, W, bias, lng, lnb, (float*)d_out);
}